// ModelBasic_38173669327415
// MI455X (gfx1250) — compile-verified
//
#include <hip/hip_runtime.h>
#include <hip/hip_bf16.h>
#include <stdint.h>

typedef float v2f __attribute__((ext_vector_type(2)));
typedef float v8f __attribute__((ext_vector_type(8)));

#define HDIM 128
#define BN_EPS 1e-5f

// LDS byte offset of a generic pointer to __shared__ (low 32 bits of LDS aperture addr)
__device__ __forceinline__ unsigned lds_off(const void* p) {
    return (unsigned)(uintptr_t)p;
}

// async global -> LDS copy of 16 bytes (CDNA5, tracked by ASYNCcnt)
__device__ __forceinline__ void async_copy_b128(const void* gptr, void* lptr) {
    unsigned l = lds_off(lptr);
    asm volatile("global_load_async_to_lds_b128 %0, %1, off"
                 :: "v"(l), "v"(gptr) : "memory");
}

__device__ __forceinline__ void async_wait0() {
    asm volatile("s_wait_asynccnt 0" ::: "memory");
}

// ---------------- utility ----------------
__global__ void fill_f32(float* __restrict__ p, size_t n, float v) {
    size_t i = (size_t)blockIdx.x * blockDim.x + threadIdx.x;
    if (i < n) p[i] = v;
}

// ---------------- degree / dinv ----------------
__global__ void degree_kernel(const int* __restrict__ dst, int E, float* __restrict__ deg) {
    int e = blockIdx.x * blockDim.x + threadIdx.x;
    if (e < E) unsafeAtomicAdd(&deg[dst[e]], 1.0f);
}

__global__ void dinv_kernel(float* __restrict__ deg, int n) {
    int i = blockIdx.x * blockDim.x + threadIdx.x;
    if (i < n) deg[i] = rsqrtf(deg[i] + 1.0f);
}

// ---------------- WMMA GEMM: H = X @ W,  X:[M,128] W:[128,128] ----------------
// block = 256 threads = 8 waves; wave w computes the 16x16 output tile at
// rows [blockIdx.x*16,+16), cols [w*16,+16).
// A tile (16x128, 8KB) staged once; W staged in two 64-row phases (32KB) via
// GLOBAL_LOAD_ASYNC_TO_LDS_B128 + s_wait_asynccnt. 32x V_WMMA_F32_16X16X4_F32.
__global__ __launch_bounds__(256) void gemm_wmma(const float* __restrict__ X,
                                                 const float* __restrict__ W,
                                                 float* __restrict__ H, int M) {
    __shared__ float Alds[16 * HDIM];     // 8 KB
    __shared__ float Wlds[64 * HDIM];     // 32 KB (one K-phase of W)

    const int lane = threadIdx.x & 31;
    const int wv   = threadIdx.x >> 5;            // col tile 0..7
    const int col  = wv * 16 + (lane & 15);
    const int mrow = lane & 15;                   // local A row
    const int khi  = (lane >> 4) << 1;            // 0 (lanes 0-15) or 2 (lanes 16-31)
    const int rt0  = blockIdx.x * 16;

    v8f acc = {};
#pragma unroll
    for (int ph = 0; ph < 2; ++ph) {
        // stage W rows [ph*64, ph*64+64): 2048 float4 chunks, 8 per thread
        for (int i = threadIdx.x; i < (64 * HDIM) / 4; i += 256) {
            async_copy_b128(W + (size_t)(ph * 64) * HDIM + (size_t)i * 4, Wlds + i * 4);
        }
        if (ph == 0) {
            // stage A tile: 512 float4 chunks, 2 per thread (rows clamped)
            for (int i = threadIdx.x; i < (16 * HDIM) / 4; i += 256) {
                int r = i >> 5;                   // (i*4)/128
                int c = (i & 31) * 4;
                int rg = rt0 + r; if (rg >= M) rg = M - 1;
                async_copy_b128(X + (size_t)rg * HDIM + c, Alds + r * HDIM + c);
            }
        }
        async_wait0();
        __syncthreads();

#pragma unroll
        for (int k = 0; k < 64; k += 4) {
            const int kk = k + khi;
            v2f a = *(const v2f*)(Alds + mrow * HDIM + ph * 64 + kk);   // ds_load_b64
            v2f b;                                                      // ds_load_b32 x2
            b.x = Wlds[kk * HDIM + col];
            b.y = Wlds[(kk + 1) * HDIM + col];
            acc = __builtin_amdgcn_wmma_f32_16x16x4_f32(false, a, false, b,
                                                        (short)0, acc, false, false);
        }
        __syncthreads();   // protect Wlds before next phase restages
    }

    const int rbase = rt0 + ((lane >> 4) << 3);   // M = j + 8*(lane>=16)
#pragma unroll
    for (int j = 0; j < 8; ++j) {
        int r = rbase + j;
        if (r < M) H[(size_t)r * HDIM + col] = acc[j];
    }
}

// ---------------- edge scatter: agg[dst] += h[src] * dinv[src]*dinv[dst] ----------------
// one wave per edge, lane handles 4 features (b128 gather, 4x global_atomic_add_f32)
__global__ __launch_bounds__(256) void scatter_kernel(const float* __restrict__ h,
                                                      const int* __restrict__ src,
                                                      const int* __restrict__ dst,
                                                      const float* __restrict__ dinv,
                                                      float* __restrict__ agg, int E) {
    int t = blockIdx.x * blockDim.x + threadIdx.x;
    int e = t >> 5;
    int lane = t & 31;
    if (e >= E) return;
    int s = src[e], d = dst[e];
    float nrm = dinv[s] * dinv[d];
    const float4 hv = *(const float4*)(h + (size_t)s * HDIM + lane * 4);
    float* ap = agg + (size_t)d * HDIM + lane * 4;
    unsafeAtomicAdd(ap + 0, hv.x * nrm);
    unsafeAtomicAdd(ap + 1, hv.y * nrm);
    unsafeAtomicAdd(ap + 2, hv.z * nrm);
    unsafeAtomicAdd(ap + 3, hv.w * nrm);
}

// ---------------- self-loop + bias ----------------
__global__ void finalize_kernel(float* __restrict__ agg, const float* __restrict__ h,
                                const float* __restrict__ dinv, const float* __restrict__ b,
                                int N) {
    size_t i = (size_t)blockIdx.x * blockDim.x + threadIdx.x;
    if (i >= (size_t)N * HDIM) return;
    int node = (int)(i >> 7);
    int f = (int)(i & (HDIM - 1));
    float di = dinv[node];
    agg[i] += h[i] * di * di + b[f];
}

// ---------------- BN column stats (partial sums + atomics) ----------------
__global__ __launch_bounds__(HDIM) void bn_stats_kernel(const float* __restrict__ agg,
                                                        float* __restrict__ sum,
                                                        float* __restrict__ sumsq,
                                                        int N, int rowsPerBlock) {
    int f = threadIdx.x;
    int r0 = blockIdx.x * rowsPerBlock;
    int r1 = r0 + rowsPerBlock; if (r1 > N) r1 = N;
    float s = 0.f, ss = 0.f;
    for (int r = r0; r < r1; ++r) {
        float v = agg[(size_t)r * HDIM + f];
        s += v; ss += v * v;
    }
    unsafeAtomicAdd(&sum[f], s);
    unsafeAtomicAdd(&sumsq[f], ss);
}

// sum -> scale, sumsq -> shift  (y = x*scale + shift)
__global__ __launch_bounds__(HDIM) void bn_coef_kernel(float* __restrict__ sum,
                                                       float* __restrict__ sumsq, int N) {
    int f = threadIdx.x;
    float inv_n = 1.0f / (float)N;
    float mean = sum[f] * inv_n;
    float var = sumsq[f] * inv_n - mean * mean;
    float sc = rsqrtf(var + BN_EPS);
    sum[f] = sc;
    sumsq[f] = -mean * sc;
}

__global__ void bn_relu_kernel(float* __restrict__ agg, const float* __restrict__ scale,
                               const float* __restrict__ shift, int N) {
    size_t i = (size_t)blockIdx.x * blockDim.x + threadIdx.x;
    if (i >= (size_t)N * HDIM) return;
    int f = (int)(i & (HDIM - 1));
    float v = fmaf(agg[i], scale[f], shift[f]);
    agg[i] = v > 0.f ? v : 0.f;
}

// ---------------- partition segment-sum with LDS staging ----------------
__global__ __launch_bounds__(256) void pe_kernel(const float* __restrict__ x,
                                                 const int* __restrict__ part,
                                                 float* __restrict__ pe,
                                                 int N, int nodesPerBlock) {
    __shared__ float lpe[16 * HDIM];
    for (int i = threadIdx.x; i < 16 * HDIM; i += 256) lpe[i] = 0.f;
    __syncthreads();
    int wv = threadIdx.x >> 5, lane = threadIdx.x & 31;
    int n0 = blockIdx.x * nodesPerBlock;
    int n1 = n0 + nodesPerBlock; if (n1 > N) n1 = N;
    for (int node = n0 + wv; node < n1; node += 8) {
        int p = part[node];
        const float4 v = *(const float4*)(x + (size_t)node * HDIM + lane * 4);
        float* pp = lpe + p * HDIM + lane * 4;      // ds_add_f32
        atomicAdd(pp + 0, v.x);
        atomicAdd(pp + 1, v.y);
        atomicAdd(pp + 2, v.z);
        atomicAdd(pp + 3, v.w);
    }
    __syncthreads();
    for (int i = threadIdx.x; i < 16 * HDIM; i += 256)
        unsafeAtomicAdd(&pe[i], lpe[i]);
}

// ---------------- head MLP (tiny: [16,256]@[256,128] -> relu -> @[128,1]) ----------------
__global__ __launch_bounds__(HDIM) void head_kernel(const float* __restrict__ x,
                                                    const float* __restrict__ pe,
                                                    const float* __restrict__ lin1W,
                                                    const float* __restrict__ lin1b,
                                                    const float* __restrict__ lin2W,
                                                    const float* __restrict__ lin2b,
                                                    const int* __restrict__ curr,
                                                    float* __restrict__ out, int P) {
    __shared__ float xc[HDIM];
    __shared__ float z[16 * HDIM];
    int j = threadIdx.x;
    int cn = curr[0];
    xc[j] = x[(size_t)cn * HDIM + j];
    __syncthreads();
    for (int p = 0; p < P; ++p) {
        float acc = lin1b[j];
        for (int k = 0; k < HDIM; ++k) acc = fmaf(xc[k], lin1W[k * HDIM + j], acc);
        for (int k = 0; k < HDIM; ++k) acc = fmaf(pe[p * HDIM + k], lin1W[(HDIM + k) * HDIM + j], acc);
        z[p * HDIM + j] = acc > 0.f ? acc : 0.f;
    }
    __syncthreads();
    if (j < P) {
        float acc = lin2b[0];
        for (int k = 0; k < HDIM; ++k) acc = fmaf(z[j * HDIM + k], lin2W[k], acc);
        out[j] = acc;
    }
}

// ---------------- host side ----------------
static inline unsigned gblk(size_t n, int b) { return (unsigned)((n + (size_t)b - 1) / (size_t)b); }

extern "C" void kernel_launch(void* const* d_in, const int* in_sizes, int n_in,
                              void* d_out, int out_size, void* d_ws, size_t ws_size,
                              hipStream_t stream) {
    const float* x     = (const float*)d_in[0];
    const float* W0    = (const float*)d_in[1];
    const float* b0    = (const float*)d_in[2];
    const float* W1    = (const float*)d_in[3];
    const float* b1    = (const float*)d_in[4];
    const float* W2    = (const float*)d_in[5];
    const float* b2    = (const float*)d_in[6];
    const float* lin1W = (const float*)d_in[7];
    const float* lin1b = (const float*)d_in[8];
    const float* lin2W = (const float*)d_in[9];
    const float* lin2b = (const float*)d_in[10];
    // d_in[11] core_values: unused by the reference
    const int* edge = (const int*)d_in[12];
    const int* curr = (const int*)d_in[13];
    const int* part = (const int*)d_in[14];

    const int N = in_sizes[0] / HDIM;
    const int E = in_sizes[12] / 2;
    const int P = out_size;          // output is [P,1] float32
    const int* srcp = edge;
    const int* dstp = edge + E;

    const size_t NH = (size_t)N * HDIM;
    float* ws = (float*)d_ws;
    float* B0   = ws;                 // h buffer
    float* B1   = ws + NH;            // agg / activation
    float* B2   = ws + 2 * NH;        // agg / activation
    float* dinv = ws + 3 * NH;        // [N]
    float* csum = dinv + N;           // [128]
    float* csq  = csum + HDIM;        // [128]
    float* pe   = csq + HDIM;         // [P*128]

    // degree + dinv (shared by all layers)
    fill_f32<<<gblk(N, 256), 256, 0, stream>>>(dinv, (size_t)N, 0.f);
    degree_kernel<<<gblk(E, 256), 256, 0, stream>>>(dstp, E, dinv);
    dinv_kernel<<<gblk(N, 256), 256, 0, stream>>>(dinv, N);

    auto run_layer = [&](const float* xin, const float* W, const float* b,
                         float* hbuf, float* aggbuf) {
        gemm_wmma<<<gblk(N, 16), 256, 0, stream>>>(xin, W, hbuf, N);
        fill_f32<<<gblk(NH, 256), 256, 0, stream>>>(aggbuf, NH, 0.f);
        scatter_kernel<<<gblk((size_t)E * 32, 256), 256, 0, stream>>>(hbuf, srcp, dstp, dinv, aggbuf, E);
        finalize_kernel<<<gblk(NH, 256), 256, 0, stream>>>(aggbuf, hbuf, dinv, b, N);
        fill_f32<<<1, 256, 0, stream>>>(csum, 2 * HDIM, 0.f);   // csum+csq contiguous
        bn_stats_kernel<<<gblk(N, 256), HDIM, 0, stream>>>(aggbuf, csum, csq, N, 256);
        bn_coef_kernel<<<1, HDIM, 0, stream>>>(csum, csq, N);
        bn_relu_kernel<<<gblk(NH, 256), 256, 0, stream>>>(aggbuf, csum, csq, N);
    };

    // 3 GCN layers with buffer rotation (BN+ReLU in place on agg buffer)
    run_layer(x,  W0, b0, B0, B1);   // x1 = B1
    run_layer(B1, W1, b1, B0, B2);   // x2 = B2
    run_layer(B2, W2, b2, B0, B1);   // x3 = B1

    // partition pooling
    fill_f32<<<gblk((size_t)P * HDIM, 256), 256, 0, stream>>>(pe, (size_t)P * HDIM, 0.f);
    pe_kernel<<<gblk(N, 512), 256, 0, stream>>>(B1, part, pe, N, 512);

    // head
    head_kernel<<<1, HDIM, 0, stream>>>(B1, pe, lin1W, lin1b, lin2W, lin2b, curr,
                                        (float*)d_out, P);
}